// TableRecognizer_7267084665426
// MI455X (gfx1250) — compile-verified
//
#include <hip/hip_runtime.h>
#include <hip/hip_bf16.h>

// ---------------------------------------------------------------------------
// Model dims (compile-time constants of the reference)
// ---------------------------------------------------------------------------
#define NB    2
#define LSEQ  1024
#define ROWS  (NB * LSEQ)      // 2048 token rows
#define DM    512
#define FFD   2048
#define HH    8
#define DK    64               // DM / HH
#define NLAY  4
#define VOC   8000
#define CIMG  512
#define NCHUNK 16              // scan chunks
#define CLEN   64              // LSEQ / NCHUNK

#define KSTEP  32              // K-panel staged per TDM transfer
#define LDS_STRIDE 80          // 64 + 16 pad DWORDs -> conflict-free banks

typedef float v2f __attribute__((ext_vector_type(2)));
typedef float v8f __attribute__((ext_vector_type(8)));
typedef unsigned int v4u __attribute__((ext_vector_type(4)));
typedef int v8i __attribute__((ext_vector_type(8)));
typedef int v4i __attribute__((ext_vector_type(4)));

// ---------------------------------------------------------------------------
// fp32 GEMM: V_WMMA_F32_16X16X4_F32 + TDM double-buffered B panels in LDS.
// Block: 128 threads = 4 waves; block tile 64(M) x 64(N); wave tile 16 x 64.
// A: MxK row-major, B: KxN row-major, C: MxN row-major.
// mode: 0 = bias, 1 = elu(x)+1, 2 = (elu(x)+1)*mask[row], 3 = exact GELU
// Requires: M % 64 == 0, N % 64 == 0, K % 32 == 0 (true for all launches).
// ---------------------------------------------------------------------------
__global__ __launch_bounds__(128) void gemm_f32_wmma(
    const float* __restrict__ A, const float* __restrict__ B,
    const float* __restrict__ bias, const float* __restrict__ mask,
    float* __restrict__ C, int M, int K, int N, int mode)
{
  __shared__ float bp[2][KSTEP * LDS_STRIDE];   // 2 x 10 KB, TDM-padded rows

  const int wave = threadIdx.x >> 5;
  const int lane = threadIdx.x & 31;
  const int nblk = N >> 6;
  const int bm = (blockIdx.x / nblk) << 6;
  const int bn = (blockIdx.x % nblk) << 6;
  const int h   = lane >> 4;                    // lane half selects K = 2h, 2h+1
  const int n16 = lane & 15;

  const float* arow = A + (size_t)(bm + wave * 16 + n16) * K + 2 * h;
  const int npanels = K >> 5;                   // >= 16 for every launch

#if __has_builtin(__builtin_amdgcn_tensor_load_to_lds)
  // TDM issue: one DMA per block (wave 0 only), D# built per ISA 8.3/8.4.
  // clang-23 toolchain: 6-arg form (g0, g1, g2, g3, g4, cpol).
  auto issue = [&](int p, int buf) {
    unsigned long long ga =
        (unsigned long long)(uintptr_t)(B + (size_t)(p << 5) * N + bn);
    unsigned int la = (unsigned int)(uintptr_t)&bp[buf][0];
    v4u g0; v8i g1; v4i g2; v4i g3; v8i g4;
    g2[0] = g2[1] = g2[2] = g2[3] = 0;
    g3[0] = g3[1] = g3[2] = g3[3] = 0;
    g4[0] = g4[1] = g4[2] = g4[3] = g4[4] = g4[5] = g4[6] = g4[7] = 0;
    g0[0] = 1u;                                          // count=1, user mode
    g0[1] = la;                                          // lds_addr
    g0[2] = (unsigned int)ga;                            // global_addr[31:0]
    g0[3] = ((unsigned int)(ga >> 32) & 0x01FFFFFFu)     // global_addr[56:32]
            | 0x80000000u;                               // type = 2 (image)
    unsigned int nd0 = (unsigned int)N;                  // tensor_dim0
    unsigned int nd1 = (unsigned int)K;                  // tensor_dim1
    g1[0] = (int)((2u << 16)                             // data_size = 4B
                | (1u << 20)                             // pad_enable
                | (5u << 22)                             // pad_interval: 64 DW
                | (15u << 25));                          // pad_amount: 16 DW
    g1[1] = (int)(nd0 << 16);                            // tensor_dim0[15:0]
    g1[2] = (int)((nd0 >> 16) | (nd1 << 16));            // dim0 hi | dim1 lo
    g1[3] = (int)((nd1 >> 16) | (64u << 16));            // dim1 hi | tile_dim0
    g1[4] = (int)(unsigned int)KSTEP;                    // tile_dim1 = 32
    g1[5] = (int)(unsigned int)N;                        // dim0_stride lo32
    g1[6] = 0;                                           // dim0_stride hi16
    g1[7] = 0;
    __builtin_amdgcn_tensor_load_to_lds(g0, g1, g2, g3, g4, 0);
  };
  if (wave == 0) {
    issue(0, 0);
    issue(1, 1);
    __builtin_amdgcn_s_wait_tensorcnt(1);   // panel 0 landed, panel 1 in flight
  }
  __syncthreads();
#else
  // Fallback: cooperative load of panel p into padded LDS.
  auto stage = [&](int p, int buf) {
    for (int i = threadIdx.x; i < KSTEP * 64; i += 128) {
      int r = i >> 6, cc = i & 63;
      bp[buf][r * LDS_STRIDE + cc] = B[(size_t)((p << 5) + r) * N + bn + cc];
    }
  };
  stage(0, 0);
  stage(1, 1);
  __syncthreads();
#endif

  v8f acc[4] = {};
  for (int p = 0; p < npanels; ++p) {
    const float* bpan = bp[p & 1];
    const int kb = p << 5;
#pragma unroll
    for (int kk = 0; kk < KSTEP; kk += 4) {
      v2f a = *(const v2f*)(arow + kb + kk);          // A[m][k..k+1], 8B load
      const int r0 = (kk + 2 * h) * LDS_STRIDE + n16; // row k = 2h
      const int r1 = r0 + LDS_STRIDE;                 // row k = 2h+1
#pragma unroll
      for (int j = 0; j < 4; ++j) {
        v2f b;
        b.x = bpan[r0 + j * 16];
        b.y = bpan[r1 + j * 16];
        acc[j] = __builtin_amdgcn_wmma_f32_16x16x4_f32(
            false, a, false, b, (short)0, acc[j], false, false);
      }
    }
    __syncthreads();                                  // panel p fully consumed
#if __has_builtin(__builtin_amdgcn_tensor_load_to_lds)
    if (wave == 0) {
      if (p + 2 < npanels) {
        issue(p + 2, p & 1);
        __builtin_amdgcn_s_wait_tensorcnt(1);         // panel p+1 landed
      } else {
        __builtin_amdgcn_s_wait_tensorcnt(0);
      }
    }
#else
    if (p + 2 < npanels) stage(p + 2, p & 1);
#endif
    __syncthreads();                                  // next panel published
  }

  // Epilogue. C/D layout: VGPR v -> row  bm + wave*16 + v + 8h, col bn+j*16+n16.
  const int row0 = bm + wave * 16 + 8 * h;
#pragma unroll
  for (int j = 0; j < 4; ++j) {
    const int col = bn + j * 16 + n16;
    const float bv = bias[col];
#pragma unroll
    for (int v = 0; v < 8; ++v) {
      const int row = row0 + v;
      float x = acc[j][v] + bv;
      if (mode == 1) {
        x = (x > 0.f) ? (x + 1.f) : expf(x);                   // elu + 1
      } else if (mode == 2) {
        float pp = (x > 0.f) ? (x + 1.f) : expf(x);
        x = pp * mask[row];
      } else if (mode == 3) {
        x = 0.5f * x * (1.f + erff(x * 0.70710678118654752f)); // exact GELU
      }
      C[(size_t)row * N + col] = x;
    }
  }
}

// ---------------------------------------------------------------------------
// x_mask = mask_img + mask_txt   (2048 values)
// ---------------------------------------------------------------------------
__global__ void mask_combine(const float* __restrict__ mi,
                             const float* __restrict__ mt,
                             float* __restrict__ xm)
{
  int i = blockIdx.x * blockDim.x + threadIdx.x;
  if (i < ROWS) xm[i] = mi[i] + mt[i];
}

// ---------------------------------------------------------------------------
// Embedding combine: x = (timg + pos_img)*mask_img + (emb_txt[idx]+pos_txt)*mask_txt
// timg = x_img @ W_ei + b_ei precomputed by GEMM.
// ---------------------------------------------------------------------------
__global__ __launch_bounds__(256) void embed_kernel(
    const float* __restrict__ timg, const int* __restrict__ x_txt,
    const float* __restrict__ emb_txt,
    const float* __restrict__ mask_img, const float* __restrict__ mask_txt,
    const float* __restrict__ pos_r, const float* __restrict__ pos_c,
    const float* __restrict__ pos_t, float* __restrict__ x)
{
  const int idx = blockIdx.x * blockDim.x + threadIdx.x;  // over ROWS*DM
  const int c  = idx & (DM - 1);
  const int rl = idx >> 9;                                // n*L + l
  const float LOG1E4 = 9.210340371976184f;

  // image positional: concat(pos_enc(pos_r,256), pos_enc(pos_c,256))
  float pe_img;
  {
    float p; int cc;
    if (c < 256) { p = pos_r[rl]; cc = c; } else { p = pos_c[rl]; cc = c - 256; }
    int j = (cc < 128) ? cc : (cc - 128);
    float ang = p * expf(-(LOG1E4 / 127.f) * (float)j);
    pe_img = (cc < 128) ? sinf(ang) : cosf(ang);
  }
  // text positional: pos_enc(pos_t, 512)
  float pe_txt;
  {
    float p = pos_t[rl];
    int j = (c < 256) ? c : (c - 256);
    float ang = p * expf(-(LOG1E4 / 255.f) * (float)j);
    pe_txt = (c < 256) ? sinf(ang) : cosf(ang);
  }
  float vi = (timg[idx] + pe_img) * mask_img[rl];
  float vt = (emb_txt[(size_t)x_txt[rl] * DM + c] + pe_txt) * mask_txt[rl];
  x[idx] = vi + vt;
}

// ---------------------------------------------------------------------------
// Fused residual-add + LayerNorm. One row (512) per block, 256 threads.
// ---------------------------------------------------------------------------
__global__ __launch_bounds__(256) void add_ln_kernel(
    const float* __restrict__ xin, const float* __restrict__ res,
    const float* __restrict__ s, const float* __restrict__ b,
    float* __restrict__ xout)
{
  __shared__ float wred[8];
  const int row = blockIdx.x;
  const int t = threadIdx.x;
  const float* xr = xin + (size_t)row * DM;
  const float* rr = res + (size_t)row * DM;
  float v0 = xr[t] + rr[t];
  float v1 = xr[t + 256] + rr[t + 256];

  float sum = v0 + v1;
  for (int o = 16; o > 0; o >>= 1) sum += __shfl_xor(sum, o, 32);
  if ((t & 31) == 0) wred[t >> 5] = sum;
  __syncthreads();
  float tot = 0.f;
#pragma unroll
  for (int w = 0; w < 8; ++w) tot += wred[w];
  const float mean = tot * (1.f / DM);

  float d0 = v0 - mean, d1 = v1 - mean;
  float vs = d0 * d0 + d1 * d1;
  for (int o = 16; o > 0; o >>= 1) vs += __shfl_xor(vs, o, 32);
  __syncthreads();
  if ((t & 31) == 0) wred[t >> 5] = vs;
  __syncthreads();
  float vtot = 0.f;
#pragma unroll
  for (int w = 0; w < 8; ++w) vtot += wred[w];
  const float inv = rsqrtf(vtot * (1.f / DM) + 1e-5f);

  xout[(size_t)row * DM + t]       = d0 * inv * s[t] + b[t];
  xout[(size_t)row * DM + t + 256] = d1 * inv * s[t + 256] + b[t + 256];
}

// ---------------------------------------------------------------------------
// Scan phase A: per-chunk sums of k and k (x) v for each (n,h,chunk).
// Thread t owns e = t&63, d in [16*(t>>6), 16*(t>>6)+16).
// ---------------------------------------------------------------------------
__global__ __launch_bounds__(256) void scan_partial(
    const float* __restrict__ Kf, const float* __restrict__ Vf,
    float* __restrict__ KVc, float* __restrict__ Ksumc)
{
  const int blk = blockIdx.x;            // = nh*NCHUNK + c
  const int c = blk % NCHUNK;
  const int nh = blk / NCHUNK;
  const int n = nh >> 3, hh = nh & 7;
  __shared__ float kl[DK], vl[DK], ks[DK];
  const int t = threadIdx.x;
  const int e = t & 63, dg = t >> 6;
  float kv[16];
#pragma unroll
  for (int i = 0; i < 16; ++i) kv[i] = 0.f;
  if (t < DK) ks[t] = 0.f;
  __syncthreads();

  for (int s2 = 0; s2 < CLEN; ++s2) {
    const int l = c * CLEN + s2;
    const size_t base = ((size_t)(n * LSEQ + l)) * DM + hh * DK;
    if (t < 64)       kl[t]       = Kf[base + t];
    else if (t < 128) vl[t - 64]  = Vf[base + t - 64];
    __syncthreads();
    const float vv = vl[e];
#pragma unroll
    for (int i = 0; i < 16; ++i) kv[i] += kl[dg * 16 + i] * vv;
    if (t < DK) ks[t] += kl[t];
    __syncthreads();
  }

  const size_t ob = (size_t)blk * (DK * DK);
#pragma unroll
  for (int i = 0; i < 16; ++i) KVc[ob + (size_t)(dg * 16 + i) * DK + e] = kv[i];
  if (t < DK) Ksumc[(size_t)blk * DK + t] = ks[t];
}

// ---------------------------------------------------------------------------
// Scan phase B: replay chunk with exclusive-prefix state, emit attention out.
// ---------------------------------------------------------------------------
__global__ __launch_bounds__(256) void scan_apply(
    const float* __restrict__ Qf, const float* __restrict__ Kf,
    const float* __restrict__ Vf, const float* __restrict__ KVc,
    const float* __restrict__ Ksumc, float* __restrict__ Of)
{
  const int blk = blockIdx.x;
  const int c = blk % NCHUNK;
  const int nh = blk / NCHUNK;
  const int n = nh >> 3, hh = nh & 7;
  __shared__ float ql[DK], kl[DK], vl[DK], ks[DK];
  __shared__ float red[4][DK];
  __shared__ float zsum;
  const int t = threadIdx.x;
  const int e = t & 63, dg = t >> 6;

  float kv[16];
#pragma unroll
  for (int i = 0; i < 16; ++i) kv[i] = 0.f;
  if (t < DK) ks[t] = 0.f;
  // exclusive prefix over earlier chunks
  for (int cp = 0; cp < c; ++cp) {
    const size_t ob = (size_t)(nh * NCHUNK + cp) * (DK * DK);
#pragma unroll
    for (int i = 0; i < 16; ++i) kv[i] += KVc[ob + (size_t)(dg * 16 + i) * DK + e];
    if (t < DK) ks[t] += Ksumc[(size_t)(nh * NCHUNK + cp) * DK + t];
  }
  __syncthreads();

  for (int s2 = 0; s2 < CLEN; ++s2) {
    const int l = c * CLEN + s2;
    const size_t base = ((size_t)(n * LSEQ + l)) * DM + hh * DK;
    if (t < 64)       ql[t]        = Qf[base + t];
    else if (t < 128) kl[t - 64]   = Kf[base + t - 64];
    else if (t < 192) vl[t - 128]  = Vf[base + t - 128];
    __syncthreads();

    // inclusive state update + partial matvec
    const float vv = vl[e];
    float po = 0.f;
#pragma unroll
    for (int i = 0; i < 16; ++i) {
      kv[i] += kl[dg * 16 + i] * vv;
      po += ql[dg * 16 + i] * kv[i];
    }
    red[dg][e] = po;
    if (t < DK) ks[t] += kl[t];
    __syncthreads();

    // denominator: dot(Q, Ksum_cum) reduced by wave 0
    if (t < 32) {
      float z = ql[t] * ks[t] + ql[t + 32] * ks[t + 32];
      for (int o = 16; o > 0; o >>= 1) z += __shfl_xor(z, o, 32);
      if (t == 0) zsum = z;
    }
    __syncthreads();

    if (t < DK) {
      float o = red[0][t] + red[1][t] + red[2][t] + red[3][t];
      Of[base + t] = o / (zsum + 1e-6f);
    }
    __syncthreads();
  }
}

// ---------------------------------------------------------------------------
// Host orchestration
// ---------------------------------------------------------------------------
extern "C" void kernel_launch(void* const* d_in, const int* in_sizes, int n_in,
                              void* d_out, int out_size, void* d_ws, size_t ws_size,
                              hipStream_t stream)
{
  (void)in_sizes; (void)n_in; (void)out_size; (void)ws_size;

  const float* x_img    = (const float*)d_in[0];
  const int*   x_txt    = (const int*)  d_in[1];
  const float* mask_img = (const float*)d_in[2];
  const float* mask_txt = (const float*)d_in[3];
  const float* pos_r    = (const float*)d_in[4];
  const float* pos_c    = (const float*)d_in[5];
  const float* pos_t    = (const float*)d_in[6];
  const float* W_ei     = (const float*)d_in[7];
  const float* b_ei     = (const float*)d_in[8];
  const float* emb_txt  = (const float*)d_in[9];
  const float* Wq  = (const float*)d_in[10];
  const float* bq  = (const float*)d_in[11];
  const float* Wk  = (const float*)d_in[12];
  const float* bk  = (const float*)d_in[13];
  const float* Wv  = (const float*)d_in[14];
  const float* bv  = (const float*)d_in[15];
  const float* Wo  = (const float*)d_in[16];
  const float* bo  = (const float*)d_in[17];
  const float* W1  = (const float*)d_in[18];
  const float* b1  = (const float*)d_in[19];
  const float* W2  = (const float*)d_in[20];
  const float* b2  = (const float*)d_in[21];
  const float* ln1_s = (const float*)d_in[22];
  const float* ln1_b = (const float*)d_in[23];
  const float* ln2_s = (const float*)d_in[24];
  const float* ln2_b = (const float*)d_in[25];
  const float* W_pi = (const float*)d_in[26];
  const float* b_pi = (const float*)d_in[27];
  const float* W_pt = (const float*)d_in[28];
  const float* b_pt = (const float*)d_in[29];

  float* out = (float*)d_out;
  float* ws  = (float*)d_ws;

  const size_t RD = (size_t)ROWS * DM;     // 1,048,576
  float* X   = ws;
  float* Y   = X  + RD;
  float* Qb  = Y  + RD;
  float* Kb  = Qb + RD;
  float* Vb  = Kb + RD;
  float* Ob  = Vb + RD;
  float* FFb = Ob + RD;                    // ROWS*FFD
  float* KVc = FFb + (size_t)ROWS * FFD;   // NB*HH*NCHUNK*64*64
  float* KSc = KVc + (size_t)NB * HH * NCHUNK * DK * DK;
  float* XM  = KSc + (size_t)NB * HH * NCHUNK * DK;

  auto gemm = [&](const float* A, const float* Bm, const float* bias,
                  const float* mask, float* Cout, int M, int K, int N, int mode) {
    int blocks = (M / 64) * (N / 64);
    gemm_f32_wmma<<<blocks, 128, 0, stream>>>(A, Bm, bias, mask, Cout, M, K, N, mode);
  };

  // x_mask = mask_img + mask_txt
  mask_combine<<<(ROWS + 255) / 256, 256, 0, stream>>>(mask_img, mask_txt, XM);

  // embedding: timg = x_img @ W_ei + b_ei  (into Qb), then combine
  gemm(x_img, W_ei, b_ei, nullptr, Qb, ROWS, CIMG, DM, 0);
  embed_kernel<<<(ROWS * DM) / 256, 256, 0, stream>>>(
      Qb, x_txt, emb_txt, mask_img, mask_txt, pos_r, pos_c, pos_t, X);

  const int scan_blocks = NB * HH * NCHUNK;   // 256
  for (int i = 0; i < NLAY; ++i) {
    const size_t wOff = (size_t)i * DM * DM;
    const size_t bOff = (size_t)i * DM;
    const size_t w1Off = (size_t)i * DM * FFD;
    const size_t b1Off = (size_t)i * FFD;

    // Q/K/V projections with fused feature map
    gemm(X, Wq + wOff, bq + bOff, nullptr, Qb, ROWS, DM, DM, 1);   // elu+1
    gemm(X, Wk + wOff, bk + bOff, XM,      Kb, ROWS, DM, DM, 2);   // (elu+1)*mask
    gemm(X, Wv + wOff, bv + bOff, nullptr, Vb, ROWS, DM, DM, 0);

    // chunked causal linear-attention scan
    scan_partial<<<scan_blocks, 256, 0, stream>>>(Kb, Vb, KVc, KSc);
    scan_apply  <<<scan_blocks, 256, 0, stream>>>(Qb, Kb, Vb, KVc, KSc, Ob);

    // output projection + residual + LN1
    gemm(Ob, Wo + wOff, bo + bOff, nullptr, Y, ROWS, DM, DM, 0);
    add_ln_kernel<<<ROWS, 256, 0, stream>>>(Y, X, ln1_s + bOff, ln1_b + bOff, X);

    // FFN with fused exact GELU, + residual + LN2
    gemm(X, W1 + w1Off, b1 + b1Off, nullptr, FFb, ROWS, DM, FFD, 3);
    gemm(FFb, W2 + w1Off, b2 + bOff, nullptr, Y, ROWS, FFD, DM, 0);
    add_ln_kernel<<<ROWS, 256, 0, stream>>>(Y, X, ln2_s + bOff, ln2_b + bOff, X);
  }

  // heads
  gemm(X, W_pi, b_pi, nullptr, out,      ROWS, DM, CIMG, 0);
  gemm(X, W_pt, b_pt, nullptr, out + RD, ROWS, DM, VOC,  0);
}